// DifferentiableSparseSoftmax_82935818485983
// MI455X (gfx1250) — compile-verified
//
#include <hip/hip_runtime.h>

#ifndef __has_builtin
#define __has_builtin(x) 0
#endif

#define NROWS  4096
#define VCOLS  32000
#define TPB    1024
#define KPT    32            // ceil(VCOLS / TPB): cols per thread (strided)
#define NWAVES (TPB / 32)    // 32 wave32s per block
#define NITER  24

#define AS1 __attribute__((address_space(1)))
#define AS3 __attribute__((address_space(3)))

typedef __attribute__((ext_vector_type(4))) int v4i;
typedef AS1 v4i* global_v4i_ptr;
typedef AS3 v4i* lds_v4i_ptr;

// ---------------- wave32 cross-lane reductions (gfx1250 is wave32-only) ----
__device__ __forceinline__ float warp_sum(float v) {
#pragma unroll
  for (int off = 16; off > 0; off >>= 1) v += __shfl_xor(v, off, 32);
  return v;
}
__device__ __forceinline__ float warp_max(float v) {
#pragma unroll
  for (int off = 16; off > 0; off >>= 1) v = fmaxf(v, __shfl_xor(v, off, 32));
  return v;
}

// Block-wide reductions; `red` has NWAVES (=32) slots, result broadcast to all.
__device__ __forceinline__ float block_sum(float v, float* red) {
  const int lane = threadIdx.x & 31;
  const int wid  = threadIdx.x >> 5;
  v = warp_sum(v);
  __syncthreads();                 // protect red[] from previous use
  if (lane == 0) red[wid] = v;
  __syncthreads();
  if (wid == 0) {
    float t = red[lane];           // exactly 32 waves -> all slots valid
    t = warp_sum(t);
    if (lane == 0) red[0] = t;
  }
  __syncthreads();
  return red[0];
}
__device__ __forceinline__ float block_max(float v, float* red) {
  const int lane = threadIdx.x & 31;
  const int wid  = threadIdx.x >> 5;
  v = warp_max(v);
  __syncthreads();
  if (lane == 0) red[wid] = v;
  __syncthreads();
  if (wid == 0) {
    float t = red[lane];
    t = warp_max(t);
    if (lane == 0) red[0] = t;
  }
  __syncthreads();
  return red[0];
}

// ---------------- per-row entmax bisection + Fenchel-Young loss ------------
__global__ __launch_bounds__(TPB)
void entmax15_loss_rows(const float* __restrict__ X,
                        const int*   __restrict__ target,
                        float*       __restrict__ row_loss) {
  extern __shared__ float lds_row[];        // VCOLS raw X values (128 KB)
  __shared__ float red[NWAVES];

  const int row = blockIdx.x;
  const float* Xrow = X + (size_t)row * VCOLS;

  // ---- stage row into LDS once: CDNA5 async global->LDS (ASYNCcnt) ----
  const int nchunks = (VCOLS * 4) / 16;     // 8000 x 16B
#if __has_builtin(__builtin_amdgcn_global_load_async_to_lds_b128)
  for (int c = threadIdx.x; c < nchunks; c += TPB) {
    __builtin_amdgcn_global_load_async_to_lds_b128(
        (global_v4i_ptr)(Xrow + c * 4),
        (lds_v4i_ptr)(lds_row + c * 4),
        /*offset=*/0, /*cpol=*/0);
  }
#if __has_builtin(__builtin_amdgcn_s_wait_asynccnt)
  __builtin_amdgcn_s_wait_asynccnt(0);
#else
  asm volatile("s_wait_asynccnt 0" ::: "memory");
#endif
#else
  for (int c = threadIdx.x * 4; c < VCOLS; c += TPB * 4) {
    float4 t = *(const float4*)(Xrow + c);
    *(float4*)(lds_row + c) = t;
  }
#endif
  __syncthreads();

  // ---- pull my 32 strided elements (as Xs = 0.5*X) into registers ----
  float r[KPT];
#pragma unroll
  for (int k = 0; k < KPT; ++k) {
    const int col = (int)threadIdx.x + k * TPB;
    r[k] = (col < VCOLS) ? 0.5f * lds_row[col] : -1.0e30f;  // pad -> relu gives 0
  }

  // ---- row max of Xs ----
  float m = -1.0e30f;
#pragma unroll
  for (int k = 0; k < KPT; ++k) m = fmaxf(m, r[k]);
  m = block_max(m, red);

  // tau bounds: tau_lo = max-1 ; tau_hi = max - (1/d)^(alpha-1) = max - 1/sqrt(32000)
  float tau_lo = m - 1.0f;
  const float tau_hi = m - 0.0055901699437494745f;

  // f_lo = sum relu(Xs - tau_lo)^2 - 1  (fixed for the whole bisection)
  float acc = 0.0f;
#pragma unroll
  for (int k = 0; k < KPT; ++k) {
    float t = fmaxf(r[k] - tau_lo, 0.0f);
    acc = fmaf(t, t, acc);
  }
  const float f_lo = block_sum(acc, red) - 1.0f;

  float dm    = tau_hi - tau_lo;
  float tau_m = tau_lo;

  // ---- 24 bisection iterations, all data in registers ----
  for (int it = 0; it < NITER; ++it) {
    dm *= 0.5f;
    tau_m = tau_lo + dm;
    acc = 0.0f;
#pragma unroll
    for (int k = 0; k < KPT; ++k) {
      float t = fmaxf(r[k] - tau_m, 0.0f);
      acc = fmaf(t, t, acc);               // relu(x)^2
    }
    const float f_m = block_sum(acc, red) - 1.0f;   // uniform across block
    if (f_m * f_lo >= 0.0f) tau_lo = tau_m;         // uniform branch
  }

  // ---- final pass at last midpoint: S = sum pm, sum pm^1.5 = t^3, sum pm*Xs
  float s_pm = 0.0f, s_pm15 = 0.0f, s_pmx = 0.0f;
#pragma unroll
  for (int k = 0; k < KPT; ++k) {
    float t  = fmaxf(r[k] - tau_m, 0.0f);
    float pm = t * t;
    s_pm  += pm;
    s_pm15 = fmaf(pm, t, s_pm15);          // pm^(3/2) = t^3
    s_pmx  = fmaf(pm, r[k], s_pmx);        // pm * Xs   (pad lanes contribute 0)
  }
  s_pm   = block_sum(s_pm,   red);
  s_pm15 = block_sum(s_pm15, red);
  s_pmx  = block_sum(s_pmx,  red);

  if (threadIdx.x == 0) {
    const float S          = s_pm;
    const float sum_palpha = s_pm15 / (S * sqrtf(S));        // S^-1.5 * sum t^3
    const float omega      = (1.0f - sum_palpha) * (4.0f / 3.0f); // 1/(a*(a-1))
    const float x_target   = lds_row[target[row]];           // raw X value
    const float dot        = (2.0f * s_pmx) / S - x_target;  // X = 2*Xs
    row_loss[row] = omega + dot;
  }
}

// ---------------- deterministic final sum over rows ------------------------
__global__ __launch_bounds__(TPB)
void reduce_rows(const float* __restrict__ row_loss, float* __restrict__ out) {
  __shared__ float red[NWAVES];
  float v = 0.0f;
  for (int i = threadIdx.x; i < NROWS; i += TPB) v += row_loss[i];
  v = block_sum(v, red);
  if (threadIdx.x == 0) out[0] = v;
}

extern "C" void kernel_launch(void* const* d_in, const int* in_sizes, int n_in,
                              void* d_out, int out_size, void* d_ws, size_t ws_size,
                              hipStream_t stream) {
  (void)in_sizes; (void)n_in; (void)out_size; (void)ws_size;
  const float* X      = (const float*)d_in[0];
  const int*   target = (const int*)d_in[1];
  float* row_loss = (float*)d_ws;           // NROWS floats of scratch
  float* out      = (float*)d_out;

  const size_t shmem = (size_t)VCOLS * sizeof(float);   // 128 KB dynamic LDS
  static int attr_done = 0;  // host-side, idempotent, not a stream op
  if (!attr_done) {
    (void)hipFuncSetAttribute((const void*)entmax15_loss_rows,
                              hipFuncAttributeMaxDynamicSharedMemorySize,
                              (int)shmem);
    attr_done = 1;
  }

  entmax15_loss_rows<<<NROWS, TPB, shmem, stream>>>(X, target, row_loss);
  reduce_rows<<<1, TPB, 0, stream>>>(row_loss, out);
}